// PointMLP_24859270709340
// MI455X (gfx1250) — compile-verified
//
#include <hip/hip_runtime.h>

// ---------------- problem constants (from reference) ----------------
constexpr int NB  = 8;       // batches
constexpr int NPT = 16384;   // points per batch
constexpr int ND  = 64;      // feature dim
constexpr int NS  = 2048;    // sampled groups (S)
constexpr int NK  = 32;      // neighbors (K)
constexpr int NC  = ND + 3;        // 67  (grouped channels)
constexpr int NCO = 2 * ND + 3;    // 131 (output channels)

typedef __attribute__((ext_vector_type(2))) float v2f;
typedef __attribute__((ext_vector_type(8))) float v8f;

// =====================================================================
// Kernel 1: farthest point sampling. One workgroup per batch.
// Point cloud held in registers (16 pts / thread); per-step argmax via
// wave32 shuffles + 32-entry LDS cross-wave reduce. 2048 serial steps.
// =====================================================================
__global__ __launch_bounds__(1024) void fps_kernel(const float* __restrict__ xyz,
                                                   int* __restrict__ fps_idx) {
    constexpr int T   = 1024;
    constexpr int PPT = NPT / T;   // 16 points per thread
    const int b = blockIdx.x;
    const int t = threadIdx.x;
    const float* xb = xyz + (size_t)b * NPT * 3;

    float px[PPT], py[PPT], pz[PPT], dist[PPT];
#pragma unroll
    for (int j = 0; j < PPT; ++j) {
        int i = j * T + t;
        px[j] = xb[i * 3 + 0];
        py[j] = xb[i * 3 + 1];
        pz[j] = xb[i * 3 + 2];
        dist[j] = 1e10f;
    }

    __shared__ float s_val[32];
    __shared__ int   s_idx[32];
    __shared__ float s_c[3];
    __shared__ int   s_far;

    int farthest = 0;
    for (int s = 0; s < NS; ++s) {
        if (t == 0) {
            fps_idx[b * NS + s] = farthest;
            s_c[0] = xb[farthest * 3 + 0];
            s_c[1] = xb[farthest * 3 + 1];
            s_c[2] = xb[farthest * 3 + 2];
        }
        __syncthreads();
        const float cx = s_c[0], cy = s_c[1], cz = s_c[2];

        float best = -1.0f;
        int   bidx = 0x7fffffff;
#pragma unroll
        for (int j = 0; j < PPT; ++j) {
            float dx = px[j] - cx, dy = py[j] - cy, dz = pz[j] - cz;
            float d  = dx * dx + dy * dy + dz * dz;
            float nd = fminf(dist[j], d);
            dist[j]  = nd;
            int i = j * T + t;
            if (nd > best || (nd == best && i < bidx)) { best = nd; bidx = i; }
        }
        // wave32 argmax reduce (tie -> lower index, matching jnp.argmax)
        for (int off = 16; off > 0; off >>= 1) {
            float ov = __shfl_xor(best, off, 32);
            int   oi = __shfl_xor(bidx, off, 32);
            if (ov > best || (ov == best && oi < bidx)) { best = ov; bidx = oi; }
        }
        if ((t & 31) == 0) { s_val[t >> 5] = best; s_idx[t >> 5] = bidx; }
        __syncthreads();
        if (t < 32) {
            best = s_val[t];
            bidx = s_idx[t];
            for (int off = 16; off > 0; off >>= 1) {
                float ov = __shfl_xor(best, off, 32);
                int   oi = __shfl_xor(bidx, off, 32);
                if (ov > best || (ov == best && oi < bidx)) { best = ov; bidx = oi; }
            }
            if (t == 0) s_far = bidx;
        }
        __syncthreads();
        farthest = s_far;
    }
}

// =====================================================================
// Kernel 2: gather sampled anchors. Also writes new_xyz into d_out.
// =====================================================================
__global__ __launch_bounds__(256) void gather_kernel(const float* __restrict__ xyz,
                                                     const float* __restrict__ points,
                                                     const int* __restrict__ fps_idx,
                                                     float* __restrict__ new_xyz,
                                                     float* __restrict__ new_pts,
                                                     float* __restrict__ out_xyz) {
    int e = blockIdx.x * 256 + threadIdx.x;   // NB*NS*NC elements
    if (e >= NB * NS * NC) return;
    int c  = e % NC;
    int bs = e / NC;
    int b  = bs / NS;
    int id = fps_idx[bs];
    if (c < ND) {
        new_pts[(size_t)bs * ND + c] = points[((size_t)b * NPT + id) * ND + c];
    } else {
        float v = xyz[((size_t)b * NPT + id) * 3 + (c - ND)];
        new_xyz[(size_t)bs * 3 + (c - ND)] = v;
        out_xyz[(size_t)bs * 3 + (c - ND)] = v;
    }
}

// =====================================================================
// Kernel 3: kNN via V_WMMA_F32_16X16X4_F32.
// One WG (8 waves / 256 thr) owns 16 queries. score = |p|^2 - 2 q.p
// (differs from sqrdist by the row-constant |q|^2 -> identical top-k).
// A row m = (qx,qy,qz,1); B col n = (-2px,-2py,-2pz,|p|^2).
// Candidates streamed in 512-wide chunks through padded LDS; per-query
// sorted top-32 lists merged serially (ties keep lower index, as top_k).
// =====================================================================
__global__ __launch_bounds__(256) void knn_kernel(const float* __restrict__ xyz,
                                                  const float* __restrict__ new_xyz,
                                                  int* __restrict__ knn_idx) {
    constexpr int QT     = 16;
    constexpr int CHUNK  = 512;
    constexpr int SPITCH = CHUNK + 4;          // bank-conflict padding
    constexpr int TILES_PER_WAVE = (CHUNK / 16) / 8;   // 4

    __shared__ float sc[QT * SPITCH];
    __shared__ float listd[QT][NK];
    __shared__ int   listi[QT][NK];
    __shared__ float q[QT][3];

    const int t    = threadIdx.x;
    const int lane = t & 31;
    const int wave = t >> 5;
    const int b    = blockIdx.x / (NS / QT);
    const int s0   = (blockIdx.x % (NS / QT)) * QT;
    const float* xb = xyz + (size_t)b * NPT * 3;

    if (t < QT * 3) q[t / 3][t % 3] = new_xyz[((size_t)b * NS + s0 + t / 3) * 3 + t % 3];
    for (int e = t; e < QT * NK; e += 256) {
        listd[e / NK][e % NK] = 3.4e38f;
        listi[e / NK][e % NK] = 0;
    }
    __syncthreads();

    // A matrix (16x4 f32, ISA layout: lanes0-15 hold K0,K1; lanes16-31 K2,K3)
    const int arow = lane & 15;
    v2f amat;
    if (lane < 16) { amat.x = q[arow][0]; amat.y = q[arow][1]; }
    else           { amat.x = q[arow][2]; amat.y = 1.0f;       }

    for (int ch = 0; ch < NPT / CHUNK; ++ch) {
        const int base = ch * CHUNK;
        if (base + CHUNK < NPT)
            __builtin_prefetch(&xb[(size_t)(base + CHUNK) * 3 + lane * 3], 0, 0);

#pragma unroll
        for (int i = 0; i < TILES_PER_WAVE; ++i) {
            const int tile = wave * TILES_PER_WAVE + i;
            const int n    = base + tile * 16 + (lane & 15);
            const float bx = xb[n * 3 + 0];
            const float by = xb[n * 3 + 1];
            const float bz = xb[n * 3 + 2];
            const float pn = bx * bx + by * by + bz * bz;
            v2f bmat;
            if (lane < 16) { bmat.x = -2.0f * bx; bmat.y = -2.0f * by; }
            else           { bmat.x = -2.0f * bz; bmat.y = pn;         }
            v8f acc = {0.f, 0.f, 0.f, 0.f, 0.f, 0.f, 0.f, 0.f};
            acc = __builtin_amdgcn_wmma_f32_16x16x4_f32(
                false, amat, false, bmat, (short)0, acc, false, false);
            // D layout: lane holds col (lane&15); VGPR v -> row v (+8 upper half)
            const int mOff = (lane < 16) ? 0 : 8;
            const int cIdx = tile * 16 + (lane & 15);
#pragma unroll
            for (int v = 0; v < 8; ++v) sc[(mOff + v) * SPITCH + cIdx] = acc[v];
        }
        __syncthreads();

        if (t < QT) {
            float* L  = listd[t];
            int*   Li = listi[t];
            for (int cI = 0; cI < CHUNK; ++cI) {
                const float val = sc[t * SPITCH + cI];
                if (val < L[NK - 1]) {
                    int j = NK - 1;
                    while (j > 0 && L[j - 1] > val) {
                        L[j] = L[j - 1]; Li[j] = Li[j - 1]; --j;
                    }
                    L[j] = val; Li[j] = base + cI;
                }
            }
        }
        __syncthreads();
    }

    for (int e = t; e < QT * NK; e += 256) {
        const int qq = e / NK, kk = e % NK;
        knn_idx[((size_t)b * NS + s0 + qq) * NK + kk] = listi[qq][kk];
    }
}

// =====================================================================
// Kernel 4: per-batch sum / sumsq of centered values (double precision),
// block-reduced then merged with global_atomic_add_f64.
// =====================================================================
constexpr int SKC = NS * NK * NC;        // 4,390,912 (divisible by 256)
constexpr int BPB = SKC / 256;           // blocks per batch = 17,152

__global__ void init_stats_kernel(double* stats) {
    if (threadIdx.x < 2 * NB) stats[threadIdx.x] = 0.0;
}

__global__ __launch_bounds__(256) void stats_kernel(const float* __restrict__ xyz,
                                                    const float* __restrict__ points,
                                                    const float* __restrict__ new_xyz,
                                                    const float* __restrict__ new_pts,
                                                    const int* __restrict__ knn_idx,
                                                    double* __restrict__ stats) {
    __shared__ double s_s[8], s_q[8];
    const int b = blockIdx.x / BPB;
    const int e = (blockIdx.x % BPB) * 256 + threadIdx.x;   // index within batch
    const int c  = e % NC;
    const int sk = e / NC;
    const int k  = sk % NK;
    const int s  = sk / NK;
    const int id = knn_idx[((size_t)b * NS + s) * NK + k];
    float v;
    if (c < ND) v = points[((size_t)b * NPT + id) * ND + c] - new_pts[((size_t)b * NS + s) * ND + c];
    else        v = xyz[((size_t)b * NPT + id) * 3 + (c - ND)] - new_xyz[((size_t)b * NS + s) * 3 + (c - ND)];

    double lv = (double)v;
    double l2 = (double)v * (double)v;
    for (int off = 16; off > 0; off >>= 1) {
        lv += __shfl_xor(lv, off, 32);
        l2 += __shfl_xor(l2, off, 32);
    }
    const int lane = threadIdx.x & 31, wave = threadIdx.x >> 5;
    if (lane == 0) { s_s[wave] = lv; s_q[wave] = l2; }
    __syncthreads();
    if (threadIdx.x == 0) {
        double a = 0.0, bq = 0.0;
        for (int w = 0; w < 8; ++w) { a += s_s[w]; bq += s_q[w]; }
        atomicAdd(&stats[b * 2 + 0], a);
        atomicAdd(&stats[b * 2 + 1], bq);
    }
}

__global__ void finalize_kernel(const double* __restrict__ stats,
                                float* __restrict__ stdinv) {
    const int b = threadIdx.x;
    if (b >= NB) return;
    const double n   = (double)SKC;
    const double sum = stats[b * 2 + 0];
    const double sq  = stats[b * 2 + 1];
    double var = (sq - sum * sum / n) / (n - 1.0);     // ddof=1
    if (var < 0.0) var = 0.0;
    const float sd = (float)sqrt(var);
    stdinv[b] = 1.0f / (sd + 1e-5f);
}

// =====================================================================
// Kernel 5: assemble output [B,S,K,131]; recompute centered from gathers.
// =====================================================================
__global__ __launch_bounds__(256) void out_kernel(const float* __restrict__ xyz,
                                                  const float* __restrict__ points,
                                                  const float* __restrict__ new_xyz,
                                                  const float* __restrict__ new_pts,
                                                  const int* __restrict__ knn_idx,
                                                  const float* __restrict__ alpha,
                                                  const float* __restrict__ beta,
                                                  const float* __restrict__ stdinv,
                                                  float* __restrict__ out) {
    const size_t total = (size_t)NB * NS * NK * NCO;
    const size_t e = (size_t)blockIdx.x * 256 + threadIdx.x;
    if (e >= total) return;
    const int    c  = (int)(e % NCO);
    const size_t sk = e / NCO;
    const int    k  = (int)(sk % NK);
    const size_t bs = sk / NK;              // b*NS + s
    const int    b  = (int)(bs / NS);
    float r;
    if (c < NC) {
        const int id = knn_idx[bs * NK + k];
        float v;
        if (c < ND) v = points[((size_t)b * NPT + id) * ND + c] - new_pts[bs * ND + c];
        else        v = xyz[((size_t)b * NPT + id) * 3 + (c - ND)] - new_xyz[bs * 3 + (c - ND)];
        r = alpha[c] * (v * stdinv[b]) + beta[c];
    } else {
        r = new_pts[bs * ND + (c - NC)];
    }
    out[e] = r;
}

// =====================================================================
// Launcher
// =====================================================================
extern "C" void kernel_launch(void* const* d_in, const int* in_sizes, int n_in,
                              void* d_out, int out_size, void* d_ws, size_t ws_size,
                              hipStream_t stream) {
    const float* xyz    = (const float*)d_in[0];   // [8,16384,3]
    const float* points = (const float*)d_in[1];   // [8,16384,64]
    const float* alpha  = (const float*)d_in[2];   // [67]
    const float* beta   = (const float*)d_in[3];   // [67]

    // d_out = new_xyz [8,2048,3] flat, then out [8,2048,32,131] flat
    float* out_xyz = (float*)d_out;
    float* out_big = (float*)d_out + (size_t)NB * NS * 3;

    // workspace layout (~6.5 MB)
    char* ws = (char*)d_ws;
    double* stats   = (double*)(ws + 0);                 // 128 B
    float*  stdinv  = (float*) (ws + 256);               // 32 B
    int*    fps_idx = (int*)   (ws + 512);               // 64 KB
    float*  new_xyz = (float*) (ws + 512 + 65536);       // 192 KB
    float*  new_pts = (float*) (ws + 512 + 65536 + 196608);             // 4 MB
    int*    knn_idx = (int*)   (ws + 512 + 65536 + 196608 + 4194304);   // 2 MB

    init_stats_kernel<<<1, 64, 0, stream>>>(stats);
    fps_kernel<<<NB, 1024, 0, stream>>>(xyz, fps_idx);

    {
        const int n = NB * NS * NC;
        gather_kernel<<<(n + 255) / 256, 256, 0, stream>>>(xyz, points, fps_idx,
                                                           new_xyz, new_pts, out_xyz);
    }

    knn_kernel<<<NB * (NS / 16), 256, 0, stream>>>(xyz, new_xyz, knn_idx);

    stats_kernel<<<NB * BPB, 256, 0, stream>>>(xyz, points, new_xyz, new_pts,
                                               knn_idx, stats);
    finalize_kernel<<<1, 32, 0, stream>>>(stats, stdinv);

    {
        const size_t total = (size_t)NB * NS * NK * NCO;
        out_kernel<<<(unsigned)((total + 255) / 256), 256, 0, stream>>>(
            xyz, points, new_xyz, new_pts, knn_idx, alpha, beta, stdinv, out_big);
    }
}